// CODESConstraintFramework_88304527606125
// MI455X (gfx1250) — compile-verified
//
#include <hip/hip_runtime.h>

// ---------------------------------------------------------------------------
// MI455X (gfx1250) implementation of the constraint-dynamics reference.
//
// Roofline: 86 GFLOP f32 GEMM (x@A, 10 steps) dominates; A (4MB) and x (17MB)
// are L2-resident (192MB), weights (134MB) are read exactly once from HBM
// (~6us @ 23.3TB/s). Rows of x evolve independently -> persistent workgroups,
// no inter-step global sync. f32 WMMA (V_WMMA_F32_16X16X4_F32) keeps exact
// dtype parity with the f32 reference. A-panels are register-double-buffered
// so L2 latency hides under the WMMA stream.
// ---------------------------------------------------------------------------

typedef __attribute__((ext_vector_type(2))) float v2f;
typedef __attribute__((ext_vector_type(8))) float v8f;

#define D        1024
#define BATCH    4096
#define NC       32
#define DT       0.01f
#define DAMPING  0.1f
#define CLAMP_V  10.0f

#define XS_STRIDE 1026   // 16 x-rows, padded: banks (2m+k)%64 distinct per half-wave
#define AP_STRIDE 18     // A panel stored transposed ap[n][k], k=0..15 + 2 pad

// ---------------------------------------------------------------------------
// Kernel 1: s[c] = sigmoid(importance)*active ; b_eff[d] = sum_c s[c]*b[c][d]
// ws layout (floats): A[0 .. D*D) | beff[D*D .. D*D+D) | s[D*D+D .. +NC)
// ---------------------------------------------------------------------------
__global__ void prep_s_beff(const float* __restrict__ importance,
                            const unsigned char* __restrict__ active,
                            const float* __restrict__ biases,
                            float* __restrict__ ws) {
  __shared__ float s_sh[NC];
  const int tid = threadIdx.x;
  if (tid < NC) {
    float sg = 1.0f / (1.0f + expf(-importance[tid]));
    s_sh[tid] = active[tid] ? sg : 0.0f;
  }
  __syncthreads();
  float be = 0.0f;
#pragma unroll
  for (int c = 0; c < NC; ++c) be += s_sh[c] * biases[c * D + tid];
  ws[(size_t)D * D + tid] = be;
  if (tid < NC) ws[(size_t)D * D + D + tid] = s_sh[tid];
}

// ---------------------------------------------------------------------------
// Kernel 2: A[i][j] = sum_c s[c]*(W[c][i][j] + W[c][j][i]).
// Row-major read is coalesced; the column gather is served out of L2 since
// all 134MB of W fits in the 192MB L2.
// ---------------------------------------------------------------------------
__global__ void build_A(const float* __restrict__ weights,
                        const float* __restrict__ ws_s,
                        float* __restrict__ A) {
  __shared__ float s_sh[NC];
  const int tid = threadIdx.x;
  if (tid < NC) s_sh[tid] = ws_s[tid];
  __syncthreads();
  const int idx = blockIdx.x * 256 + tid;
  const int i = idx >> 10;
  const int j = idx & (D - 1);
  float acc = 0.0f;
#pragma unroll 4
  for (int c = 0; c < NC; ++c) {
    const size_t base = (size_t)c * D * D;
    acc += s_sh[c] * (weights[base + (size_t)i * D + j] +
                      weights[base + (size_t)j * D + i]);
  }
  A[idx] = acc;
}

// ---------------------------------------------------------------------------
// Kernel 3: persistent dynamics. One WG = 16 batch rows, 8 waves; wave w owns
// columns [w*128, w*128+128). x tile in LDS, v in registers, A streamed from
// L2 through an LDS panel (transposed so B-fragments are single b64 loads).
// Panel p+1's global loads are issued into registers before computing panel p
// (register double-buffer) so L2 latency overlaps the WMMA stream.
// ---------------------------------------------------------------------------
__global__ void __launch_bounds__(256, 1)
dyn_step(const float* __restrict__ x_in,
         const float* __restrict__ A,
         const float* __restrict__ beff,
         const int*   __restrict__ num_steps,
         float* __restrict__ x_out) {
  __shared__ float xs[16 * XS_STRIDE];     //  65,664 B
  __shared__ float ap[D * AP_STRIDE];      //  73,728 B  (139 KB total < 320 KB)

  const int tid  = threadIdx.x;
  const int wave = tid >> 5;               // 0..7
  const int lane = tid & 31;
  const int hf   = lane >> 4;              // half-wave select (K/M offset)
  const int lrow = lane & 15;
  const int r0   = blockIdx.x * 16;        // batch-row base
  const int n0   = wave * 128;             // column base for this wave

  // Stage x tile (coalesced global -> LDS)
#pragma unroll 8
  for (int i = 0; i < 64; ++i) {
    const int flat = i * 256 + tid;
    const int m = flat >> 10, n = flat & (D - 1);
    xs[m * XS_STRIDE + n] = x_in[(size_t)(r0 + m) * D + n];
  }

  // Per-lane b_eff for the 8 column tiles this lane touches
  float breg[8];
#pragma unroll
  for (int t = 0; t < 8; ++t) breg[t] = beff[n0 + t * 16 + lrow];

  // Velocity: 8 tiles x v8f, matches WMMA C/D layout (elem j <-> M=j+8*hf)
  v8f vvel[8];
#pragma unroll
  for (int t = 0; t < 8; ++t)
#pragma unroll
    for (int j = 0; j < 8; ++j) vvel[t][j] = 0.0f;

  const int ns = num_steps[0];

  // Per-thread panel-copy mapping: flat = i*256+tid ; k = flat>>10 ; n = flat&1023
  float pref[64];  // register double-buffer for the next A panel (fully unrolled)

  for (int step = 0; step < ns; ++step) {
    v8f acc[8];
#pragma unroll
    for (int t = 0; t < 8; ++t)
#pragma unroll
      for (int j = 0; j < 8; ++j) acc[t][j] = 0.0f;

    // Preload panel 0 of this step into registers
#pragma unroll
    for (int i = 0; i < 64; ++i) {
      const int flat = i * 256 + tid;
      const int k = flat >> 10, n = flat & (D - 1);
      pref[i] = A[(size_t)k * D + n];
    }

    // K loop over 64 panels of 16 rows of A
    for (int p = 0; p < 64; ++p) {
      __syncthreads();                     // previous panel consumed; ap writable
      // Scatter staged panel p into LDS transposed: ap[n*18 + k] = A[kb+k][n]
#pragma unroll
      for (int i = 0; i < 64; ++i) {
        const int flat = i * 256 + tid;
        const int k = flat >> 10, n = flat & (D - 1);
        ap[n * AP_STRIDE + k] = pref[i];
      }
      __syncthreads();                     // panel p visible to all waves

      // Issue panel p+1's global loads now; they complete under the WMMAs
      if (p < 63) {
        const int kb2 = (p + 1) * 16;
#pragma unroll
        for (int i = 0; i < 64; ++i) {
          const int flat = i * 256 + tid;
          const int k = flat >> 10, n = flat & (D - 1);
          pref[i] = A[(size_t)(kb2 + k) * D + n];
        }
      }

      const int kb = p * 16;
#pragma unroll
      for (int kc = 0; kc < 16; kc += 4) {
        // A-frag: lane L holds {x[L%16][k0+2*hf], x[L%16][k0+2*hf+1]}
        const v2f afrag =
            *(const v2f*)&xs[lrow * XS_STRIDE + kb + kc + hf * 2];
#pragma unroll
        for (int t = 0; t < 8; ++t) {
          // B-frag: lane L holds {A[k0+2*hf][n], A[k0+2*hf+1][n]}
          const v2f bfrag =
              *(const v2f*)&ap[(n0 + t * 16 + lrow) * AP_STRIDE + kc + hf * 2];
          acc[t] = __builtin_amdgcn_wmma_f32_16x16x4_f32(
              false, afrag, false, bfrag, (short)0, acc[t], false, false);
        }
      }
    }
    __syncthreads();   // all waves done reading xs for this step

    // v' = 0.9 v - (xA + b)*dt ; x' = clamp(x + v' dt)
#pragma unroll
    for (int t = 0; t < 8; ++t) {
      const int n = n0 + t * 16 + lrow;
#pragma unroll
      for (int j = 0; j < 8; ++j) {
        const int m = j + hf * 8;
        const float f  = -(acc[t][j] + breg[t]);
        const float vv = vvel[t][j] * (1.0f - DAMPING) + f * DT;
        vvel[t][j] = vv;
        float xn = xs[m * XS_STRIDE + n] + vv * DT;
        xn = fminf(fmaxf(xn, -CLAMP_V), CLAMP_V);
        xs[m * XS_STRIDE + n] = xn;   // disjoint per-wave column slices
      }
    }
    // next step's first __syncthreads() orders these writes vs. matmul reads
  }

  __syncthreads();
#pragma unroll 8
  for (int i = 0; i < 64; ++i) {
    const int flat = i * 256 + tid;
    const int m = flat >> 10, n = flat & (D - 1);
    x_out[(size_t)(r0 + m) * D + n] = xs[m * XS_STRIDE + n];
  }
}

// ---------------------------------------------------------------------------
extern "C" void kernel_launch(void* const* d_in, const int* in_sizes, int n_in,
                              void* d_out, int out_size, void* d_ws, size_t ws_size,
                              hipStream_t stream) {
  const float*         state      = (const float*)d_in[0];          // [4096,1024]
  const float*         weights    = (const float*)d_in[1];          // [32,1024,1024]
  const float*         biases     = (const float*)d_in[2];          // [32,1024]
  const float*         importance = (const float*)d_in[3];          // [32]
  const unsigned char* active     = (const unsigned char*)d_in[4];  // [32] bool
  const int*           num_steps  = (const int*)d_in[5];            // [1]
  float* out = (float*)d_out;

  float* Aws  = (float*)d_ws;                 // D*D floats
  float* beff = Aws + (size_t)D * D;          // D floats
  float* sws  = beff + D;                     // NC floats

  prep_s_beff<<<dim3(1), dim3(D), 0, stream>>>(importance, active, biases,
                                               (float*)d_ws);
  build_A<<<dim3((D * D) / 256), dim3(256), 0, stream>>>(weights, sws, Aws);
  dyn_step<<<dim3(BATCH / 16), dim3(256), 0, stream>>>(state, Aws, beff,
                                                       num_steps, out);
}